// CrowdCountingLoss_86337432584637
// MI455X (gfx1250) — compile-verified
//
#include <hip/hip_runtime.h>
#include <math.h>

#define N       2304
#define GRID_W  48
#define BATCH   8
#define NSLOT   32

typedef float v2f __attribute__((ext_vector_type(2)));
typedef float v8f __attribute__((ext_vector_type(8)));

// ---------------------------------------------------------------------------
// prep: per (tensor, batch) sum; a = x/(sum+1e-12); log_a = log(a+1e-20)
// ---------------------------------------------------------------------------
__global__ void prep_kernel(const float* __restrict__ pred,
                            const float* __restrict__ gt,
                            float* __restrict__ a_buf,    // [2][8][N]
                            float* __restrict__ loga_buf, // [2][8][N]
                            float* __restrict__ sums)     // [2][8]
{
    __shared__ float sred[256];
    int wg = blockIdx.x;                 // 0..15
    int t = wg >> 3, b = wg & 7;
    const float* x = (t == 0 ? pred : gt) + b * N;
    int tid = threadIdx.x;
    float loc[9];
    float s = 0.f;
    for (int r = 0; r < 9; ++r) { loc[r] = x[tid + r * 256]; s += loc[r]; }
    sred[tid] = s; __syncthreads();
    for (int off = 128; off > 0; off >>= 1) {
        if (tid < off) sred[tid] += sred[tid + off];
        __syncthreads();
    }
    float S = sred[0];
    if (tid == 0) sums[wg] = S;
    float inv = 1.0f / (S + 1e-12f);
    float* ab = a_buf + wg * N;
    float* lb = loga_buf + wg * N;
    for (int r = 0; r < 9; ++r) {
        float av = loc[r] * inv;
        ab[tid + r * 256] = av;
        lb[tid + r * 256] = logf(av + 1e-20f);
    }
}

__global__ void zero_kernel(float* __restrict__ p, int n)
{
    int i = blockIdx.x * blockDim.x + threadIdx.x;
    if (i < n) p[i] = 0.f;
}

// ---------------------------------------------------------------------------
// build_v: per slot s = kind*8 + batch:
//   u_j = logw_j + h_j/eps ; mu = max_j u_j ; V[s][j] = exp(u_j - mu)
// kind 0: (log_b, g) -> ft ; kind 1: (log_a, f) -> gt
// kind 2: (log_a, p_a)     ; kind 3: (log_b, p_b)
// ---------------------------------------------------------------------------
__global__ void build_v_kernel(const float* __restrict__ loga_buf, // [2][8][N]
                               const float* __restrict__ pot_in,   // [4][8][N]
                               float* __restrict__ V,              // [32][N]
                               float* __restrict__ mu,             // [32]
                               float inv_eps)
{
    __shared__ float sred[256];
    int s = blockIdx.x;
    int kind = s >> 3, b = s & 7;
    int t  = (kind == 0 || kind == 3) ? 1 : 0;   // which log-measure
    int hk = (kind < 2) ? (1 - kind) : kind;     // which potential feeds h
    const float* lw = loga_buf + (t * BATCH + b) * N;
    const float* h  = pot_in   + (hk * BATCH + b) * N;
    int tid = threadIdx.x;
    float u[9]; float m = -3.4e38f;
    for (int r = 0; r < 9; ++r) {
        int j = tid + r * 256;
        u[r] = lw[j] + h[j] * inv_eps;
        m = fmaxf(m, u[r]);
    }
    sred[tid] = m; __syncthreads();
    for (int off = 128; off > 0; off >>= 1) {
        if (tid < off) sred[tid] = fmaxf(sred[tid], sred[tid + off]);
        __syncthreads();
    }
    float M = sred[0];
    if (tid == 0) mu[s] = M;
    float* vs = V + s * N;
    for (int r = 0; r < 9; ++r)
        vs[tid + r * 256] = __expf(u[r] - M);
}

// ---------------------------------------------------------------------------
// softmin GEMM:  S[i][s] = sum_j exp(-C_ij/eps) * V[s][j]  via f32 WMMA,
// K entries generated in-register from grid coordinates (float-incremental,
// exactly periodic xj -> no int/float converts in the hot loop).
// WG = 256 thr = 8 waves; wave w: row-tile rt = w&1 (16 rows),
// k-quarter kq = w>>1 (576 of the j-reduction).
// LDS combines the 4 k-quarter partials, epilogue does the Sinkhorn update.
// mode 0: averaged scan update (all kinds)
// mode 1: f <- softmin (replace), others copy
// mode 2: g <- softmin (replace), others copy
// ---------------------------------------------------------------------------
__launch_bounds__(256)
__global__ void softmin_wmma_kernel(const float* __restrict__ V,      // [32][N]
                                    const float* __restrict__ mu,     // [32]
                                    const float* __restrict__ pot_in, // [4][8][N]
                                    float* __restrict__ pot_out,      // [4][8][N]
                                    float eps, int mode)
{
    __shared__ float red[6][32][16];
    const float scale = -0.5f / eps;
    const float cell  = 224.0f / GRID_W;
    const float xmax  = 224.0f;          // GRID_W * cell
    int tid = threadIdx.x;
    int w = tid >> 5, lane = tid & 31;
    int rt = w & 1, kq = w >> 1;
    int rowbase = blockIdx.x * 32 + rt * 16;

    // A-matrix (16x4 f32) layout: lane l holds row M=l&15,
    // VGPR0 = K(kkb), VGPR1 = K(kkb+1), kkb = (l>>4)*2
    int arow = rowbase + (lane & 15);
    float yi = ((float)(arow / GRID_W) + 0.5f) * cell;
    float xi = ((float)(arow % GRID_W) + 0.5f) * cell;
    int kkb = (lane >> 4) * 2;

    int k0 = kq * (N / 4);               // 576 per quarter = 12 grid rows
    int k1 = k0 + (N / 4);
    // j = k + kkb starts at column kkb (since 576 % 48 == 0), row kq*12
    const float xj0 = ((float)kkb + 0.5f) * cell;   // exact cycle restart
    float xj = xj0;
    float yj = ((float)(kq * 12) + 0.5f) * cell;

    v8f c0 = {}; v8f c1 = {};
    for (int k = k0; k < k1; k += 4) {
        // j is even -> j+1 never wraps a grid row: yj1 = yj, xj1 = xj + cell
        float dx0 = xi - xj;
        float dx1 = dx0 - cell;
        float dy  = yi - yj;
        float dyy = dy * dy;
        v2f a;
        a.x = __expf(scale * fmaf(dx0, dx0, dyy));
        a.y = __expf(scale * fmaf(dx1, dx1, dyy));

        // B-matrix (4x16 f32): lane l holds col N=l&15, rows kkb / kkb+1
        int col = lane & 15;
        v2f b0 = *(const v2f*)(V + (col)      * N + k + kkb);
        v2f b1 = *(const v2f*)(V + (col + 16) * N + k + kkb);

        c0 = __builtin_amdgcn_wmma_f32_16x16x4_f32(false, a, false, b0,
                                                   (short)0, c0, false, false);
        c1 = __builtin_amdgcn_wmma_f32_16x16x4_f32(false, a, false, b1,
                                                   (short)0, c1, false, false);

        xj += 4.0f * cell;
        if (xj >= xmax) { xj = xj0; yj += cell; }
    }

    if (kq > 0) {
        int wi = w - 2;
        for (int e = 0; e < 8; ++e) red[wi][lane][e]     = c0[e];
        for (int e = 0; e < 8; ++e) red[wi][lane][e + 8] = c1[e];
    }
    __syncthreads();
    if (kq == 0) {
        for (int p = 0; p < 3; ++p) {
            int wi = w + 2 * p;          // partials from waves w+2, w+4, w+6
            for (int e = 0; e < 8; ++e) {
                c0[e] += red[wi][lane][e];
                c1[e] += red[wi][lane][e + 8];
            }
        }
        // C/D layout: lane l, vgpr e -> (M = e + 8*(l>>4), N = l&15)
        int col  = lane & 15;
        int moff = (lane >> 4) * 8;
        for (int t = 0; t < 2; ++t) {
            int s = t * 16 + col;
            int kind = s >> 3, b = s & 7;
            float msl = mu[s];
            const float* pin = pot_in  + (kind * BATCH + b) * N;
            float*       pou = pot_out + (kind * BATCH + b) * N;
            for (int e = 0; e < 8; ++e) {
                float Sv = (t == 0) ? c0[e] : c1[e];
                int i = rowbase + moff + e;
                float val = -eps * (logf(Sv + 1e-38f) + msl);
                float old = pin[i];
                float o;
                if (mode == 0)      o = 0.5f * (old + val);
                else if (mode == 1) o = (kind == 0) ? val : old;
                else                o = (kind == 1) ? val : old;
                pou[i] = o;
            }
        }
    }
}

// ---------------------------------------------------------------------------
// finalize: spatial = mean_b sum_i (a f + b g - a pa - b pb);
// count = mean_b (sum_pred - sum_gt)^2; loss = 0.1*spatial + 0.9*count
// ---------------------------------------------------------------------------
__global__ void finalize_kernel(const float* __restrict__ a_buf, // [2][8][N]
                                const float* __restrict__ pot,   // [4][8][N]
                                const float* __restrict__ sums,  // [2][8]
                                float* __restrict__ out)
{
    __shared__ float sred[256];
    int tid = threadIdx.x;
    float acc = 0.f;
    for (int b = 0; b < BATCH; ++b) {
        const float* a  = a_buf + b * N;
        const float* bb = a_buf + (BATCH + b) * N;
        const float* f  = pot + (0 * BATCH + b) * N;
        const float* g  = pot + (1 * BATCH + b) * N;
        const float* pa = pot + (2 * BATCH + b) * N;
        const float* pb = pot + (3 * BATCH + b) * N;
        float part = 0.f;
        for (int i = tid; i < N; i += 256)
            part += a[i] * f[i] + bb[i] * g[i] - a[i] * pa[i] - bb[i] * pb[i];
        sred[tid] = part; __syncthreads();
        for (int off = 128; off > 0; off >>= 1) {
            if (tid < off) sred[tid] += sred[tid + off];
            __syncthreads();
        }
        if (tid == 0) acc += sred[0];
        __syncthreads();
    }
    if (tid == 0) {
        float cl = 0.f;
        for (int b = 0; b < BATCH; ++b) {
            float d = sums[b] - sums[BATCH + b];
            cl += d * d;
        }
        cl *= (1.0f / BATCH);
        float spatial = acc * (1.0f / BATCH);
        out[0] = 0.1f * spatial + 0.9f * cl;
    }
}

// ---------------------------------------------------------------------------
extern "C" void kernel_launch(void* const* d_in, const int* in_sizes, int n_in,
                              void* d_out, int out_size, void* d_ws, size_t ws_size,
                              hipStream_t stream)
{
    (void)in_sizes; (void)n_in; (void)out_size; (void)ws_size;
    const float* pred = (const float*)d_in[0];
    const float* gt   = (const float*)d_in[1];
    float* out = (float*)d_out;

    float* ws = (float*)d_ws;
    size_t off = 0;
    float* a_buf = ws + off; off += 2 * BATCH * N;
    float* loga  = ws + off; off += 2 * BATCH * N;
    float* sums  = ws + off; off += 2 * BATCH;
    float* pot0  = ws + off; off += 4 * BATCH * N;
    float* pot1  = ws + off; off += 4 * BATCH * N;
    float* V     = ws + off; off += NSLOT * N;
    float* mu    = ws + off; off += NSLOT;

    prep_kernel<<<16, 256, 0, stream>>>(pred, gt, a_buf, loga, sums);
    int potn = 4 * BATCH * N;
    zero_kernel<<<(potn + 255) / 256, 256, 0, stream>>>(pot0, potn);

    // geomloss-style epsilon schedule (static)
    float eps_list[64]; int ne = 0;
    double s = 224.0;
    while (s > 0.05 && ne < 63) { float sf = (float)s; eps_list[ne++] = sf * sf; s *= 0.8; }
    { float sf = 0.05f; eps_list[ne++] = sf * sf; }

    float* pin = pot0; float* pou = pot1;
    for (int r = 0; r < ne; ++r) {
        float eps = eps_list[r];
        build_v_kernel<<<NSLOT, 256, 0, stream>>>(loga, pin, V, mu, 1.0f / eps);
        softmin_wmma_kernel<<<N / 32, 256, 0, stream>>>(V, mu, pin, pou, eps, 0);
        float* tmp = pin; pin = pou; pou = tmp;
    }
    float epsl = eps_list[ne - 1];
    for (int mode = 1; mode <= 2; ++mode) {
        build_v_kernel<<<NSLOT, 256, 0, stream>>>(loga, pin, V, mu, 1.0f / epsl);
        softmin_wmma_kernel<<<N / 32, 256, 0, stream>>>(V, mu, pin, pou, epsl, mode);
        float* tmp = pin; pin = pou; pou = tmp;
    }
    finalize_kernel<<<1, 256, 0, stream>>>(a_buf, pin, sums, out);
}